// TopoGNN_8332236554373
// MI455X (gfx1250) — compile-verified
//
#include <hip/hip_runtime.h>
#include <hip/hip_bf16.h>

typedef __attribute__((ext_vector_type(2))) float v2f;
typedef __attribute__((ext_vector_type(8))) float v8f;

#define HID 64
#define NB 64
#define EPS_BN 1e-5f

// ---------------------------------------------------------------------------
// degree / scatter kernels (segment sums)
// ---------------------------------------------------------------------------
__global__ void degree_kernel(const int* __restrict__ dst, float* __restrict__ deg, int E) {
    int e = blockIdx.x * blockDim.x + threadIdx.x;
    if (e >= E) return;
    atomicAdd(&deg[dst[e]], 1.0f);
}

// pad x (N x 6) into xp (N x 8, cols 6..7 = 0)
__global__ void pad6to8(const float* __restrict__ x, float* __restrict__ xp, int n) {
    int i = blockIdx.x * blockDim.x + threadIdx.x;
    if (i >= n * 8) return;
    int node = i >> 3, c = i & 7;
    xp[i] = (c < 6) ? x[(long long)node * 6 + c] : 0.0f;
}

// scatter x (stride 6) into agg with row stride 8 (cols 6..7 stay zero)
__global__ void scatter_sum6p(const float* __restrict__ x, const int* __restrict__ src,
                              const int* __restrict__ dst, float* __restrict__ agg, int E) {
    int e = blockIdx.x * blockDim.x + threadIdx.x;
    if (e >= E) return;
    int s = src[e], d = dst[e];
    const float* xs = x + (long long)s * 6;
    float* ad = agg + (long long)d * 8;
#pragma unroll
    for (int c = 0; c < 6; ++c) atomicAdd(&ad[c], xs[c]);
}

__global__ void scatter_sum64(const float* __restrict__ h, const int* __restrict__ src,
                              const int* __restrict__ dst, float* __restrict__ agg, long long total) {
    long long idx = (long long)blockIdx.x * blockDim.x + threadIdx.x;
    if (idx >= total) return;
    int e  = (int)(idx >> 4);
    int cg = ((int)idx & 15) * 4;
    int s = src[e], d = dst[e];
    const float4 v = *(const float4*)(h + (long long)s * HID + cg);
    float* ad = agg + (long long)d * HID + cg;
    atomicAdd(ad + 0, v.x);
    atomicAdd(ad + 1, v.y);
    atomicAdd(ad + 2, v.z);
    atomicAdd(ad + 3, v.w);
}

// ---------------------------------------------------------------------------
// Fused SAGE GEMM via V_WMMA_F32_16X16X4_F32 (wave32):
//   hout[16x64 tile] = (agg/deg) @ Wl + xin @ Wr + bias   (pre-BN)
// K = padded inner dim (row stride of agg/xin, multiple of 4, zero-padded),
// KIN = valid weight rows. One wave per 16-row tile, 8 waves/block.
// A/X tiles staged to LDS via coalesced b128 loads; LDS row strides padded
// for bank-conflict-free fragment reads. Per-channel BN sums reduced in LDS.
// ---------------------------------------------------------------------------
template <int K, int KIN>
__global__ __launch_bounds__(256) void sage_gemm_wmma(
    const float* __restrict__ agg, const float* __restrict__ xin,
    const float* __restrict__ deg,
    const float* __restrict__ Wl, const float* __restrict__ Wr,
    const float* __restrict__ bias,
    float* __restrict__ hout, float* __restrict__ stats, int nnodes) {
    constexpr int WS = HID + 1;  // weight LDS row stride (bank spread)
    constexpr int AS = K + 4;    // tile LDS row stride: (4r+k) mod 64 distinct
    __shared__ float lWl[K * WS];
    __shared__ float lWr[K * WS];
    __shared__ float lb[HID];
    __shared__ float lsum[HID];
    __shared__ float lsq[HID];
    __shared__ float lA[8][16 * AS];
    __shared__ float lX[8][16 * AS];

    const int tid = threadIdx.x;
    for (int i = tid; i < K * HID; i += 256) {
        int k = i >> 6, c = i & 63;
        lWl[k * WS + c] = (k < KIN) ? Wl[k * HID + c] : 0.0f;
        lWr[k * WS + c] = (k < KIN) ? Wr[k * HID + c] : 0.0f;
    }
    if (tid < HID) {
        lb[tid]   = bias[tid];
        lsum[tid] = 0.0f;
        lsq[tid]  = 0.0f;
    }
    __syncthreads();

    const int wave = tid >> 5;
    const int lane = tid & 31;
    const int tile = blockIdx.x * 8 + wave;
    const int ntiles = nnodes >> 4;

    if (tile < ntiles) {
        // ---- stage this wave's 16xK tiles into LDS (coalesced float4) ----
        const float* gA = agg + (long long)tile * 16 * K;
        const float* gX = xin + (long long)tile * 16 * K;
        float* sA = lA[wave];
        float* sX = lX[wave];
#pragma unroll
        for (int i = 0; i < (16 * K) / 128; ++i) {
            int f = (lane + 32 * i) * 4;  // float index in tile
            int r = f / K, c = f % K;
            float4 va = *(const float4*)(gA + f);
            float4 vx = *(const float4*)(gX + f);
            *(float4*)&sA[r * AS + c] = va;
            *(float4*)&sX[r * AS + c] = vx;
        }

        const int rl  = lane & 15;  // row within tile / column within col-tile
        const int kh  = lane >> 4;  // 0: K=k0,k0+1 ; 1: K=k0+2,k0+3
        const int row = tile * 16 + rl;

        const float inv = 1.0f / fmaxf(deg[row], 1.0f);
        const float* tA = &sA[rl * AS];
        const float* tX = &sX[rl * AS];

        v8f acc[4] = {};  // 4 column tiles of 16

#pragma unroll
        for (int k0 = 0; k0 < K; k0 += 4) {
            const int ka = k0 + 2 * kh;
            v2f al, ar;
            al.x = tA[ka] * inv;
            al.y = tA[ka + 1] * inv;
            ar.x = tX[ka];
            ar.y = tX[ka + 1];
#pragma unroll
            for (int t = 0; t < 4; ++t) {
                const int col = t * 16 + rl;
                v2f bl, br;
                bl.x = lWl[ka * WS + col];
                bl.y = lWl[(ka + 1) * WS + col];
                br.x = lWr[ka * WS + col];
                br.y = lWr[(ka + 1) * WS + col];
                acc[t] = __builtin_amdgcn_wmma_f32_16x16x4_f32(
                    false, al, false, bl, (short)0, acc[t], false, false);
                acc[t] = __builtin_amdgcn_wmma_f32_16x16x4_f32(
                    false, ar, false, br, (short)0, acc[t], false, false);
            }
        }

        // D layout: VGPR j, lanes 0-15 -> row j ; lanes 16-31 -> row j+8.
#pragma unroll
        for (int t = 0; t < 4; ++t) {
            const int col = t * 16 + rl;
            const float bv = lb[col];
            float ps = 0.0f, pq = 0.0f;
#pragma unroll
            for (int j = 0; j < 8; ++j) {
                float v = acc[t][j] + bv;
                int r = tile * 16 + j + 8 * kh;
                hout[(long long)r * HID + col] = v;
                ps += v;
                pq += v * v;
            }
            atomicAdd(&lsum[col], ps);
            atomicAdd(&lsq[col], pq);
        }
    }
    __syncthreads();
    if (tid < HID) {
        atomicAdd(&stats[tid], lsum[tid]);
        atomicAdd(&stats[HID + tid], lsq[tid]);
    }
}

// ---------------------------------------------------------------------------
// BatchNorm (population stats) + ReLU, in place
// ---------------------------------------------------------------------------
__global__ void bn_relu_kernel(float* __restrict__ h, const float* __restrict__ stats,
                               const float* __restrict__ g, const float* __restrict__ be, int n) {
    long long idx = (long long)blockIdx.x * blockDim.x + threadIdx.x;
    if (idx >= (long long)n * HID) return;
    int c = (int)(idx & 63);
    float invN = 1.0f / (float)n;
    float mu  = stats[c] * invN;
    float var = stats[HID + c] * invN - mu * mu;
    float sc  = rsqrtf(var + EPS_BN) * g[c];
    float sh  = be[c] - mu * sc;
    float v   = h[idx] * sc + sh;
    h[idx] = v > 0.0f ? v : 0.0f;
}

// ---------------------------------------------------------------------------
// Softmax over x[:,4]  (the "-min" and "+1e-8" terms cancel inside softmax)
// ---------------------------------------------------------------------------
__global__ void reduce_max_part(const float* __restrict__ x, int n, float* __restrict__ part) {
    __shared__ float s[256];
    float m = -3.402823466e38f;
    for (int i = blockIdx.x * 256 + threadIdx.x; i < n; i += 256 * 256)
        m = fmaxf(m, x[(long long)i * 6 + 4]);
    s[threadIdx.x] = m;
    __syncthreads();
    for (int w = 128; w > 0; w >>= 1) {
        if (threadIdx.x < w) s[threadIdx.x] = fmaxf(s[threadIdx.x], s[threadIdx.x + w]);
        __syncthreads();
    }
    if (threadIdx.x == 0) part[blockIdx.x] = s[0];
}

__global__ void reduce_max_final(const float* __restrict__ part, float* __restrict__ md) {
    __shared__ float s[256];
    s[threadIdx.x] = part[threadIdx.x];
    __syncthreads();
    for (int w = 128; w > 0; w >>= 1) {
        if (threadIdx.x < w) s[threadIdx.x] = fmaxf(s[threadIdx.x], s[threadIdx.x + w]);
        __syncthreads();
    }
    if (threadIdx.x == 0) md[0] = s[0];
}

__global__ void reduce_sumexp_part(const float* __restrict__ x, const float* __restrict__ md,
                                   int n, float* __restrict__ part) {
    __shared__ float s[256];
    float mx = md[0];
    float a = 0.0f;
    for (int i = blockIdx.x * 256 + threadIdx.x; i < n; i += 256 * 256)
        a += __expf(x[(long long)i * 6 + 4] - mx);
    s[threadIdx.x] = a;
    __syncthreads();
    for (int w = 128; w > 0; w >>= 1) {
        if (threadIdx.x < w) s[threadIdx.x] += s[threadIdx.x + w];
        __syncthreads();
    }
    if (threadIdx.x == 0) part[blockIdx.x] = s[0];
}

__global__ void reduce_sumexp_final(const float* __restrict__ part, float* __restrict__ md) {
    __shared__ float s[256];
    s[threadIdx.x] = part[threadIdx.x];
    __syncthreads();
    for (int w = 128; w > 0; w >>= 1) {
        if (threadIdx.x < w) s[threadIdx.x] += s[threadIdx.x + w];
        __syncthreads();
    }
    if (threadIdx.x == 0) md[1] = s[0];
}

// ---------------------------------------------------------------------------
// Attention-weighted batch pooling
// ---------------------------------------------------------------------------
__global__ void bcnt_kernel(const int* __restrict__ batch, float* __restrict__ bcnt, int n) {
    int i = blockIdx.x * blockDim.x + threadIdx.x;
    if (i >= n) return;
    atomicAdd(&bcnt[batch[i]], 1.0f);
}

__global__ void pool_scatter(const float* __restrict__ h, const float* __restrict__ x,
                             const int* __restrict__ batch, const float* __restrict__ md,
                             float* __restrict__ pool_s, int n) {
    long long idx = (long long)blockIdx.x * blockDim.x + threadIdx.x;
    if (idx >= (long long)n * 16) return;
    int node = (int)(idx >> 4);
    int cg   = ((int)idx & 15) * 4;
    float w = __expf(x[(long long)node * 6 + 4] - md[0]) / md[1];
    int b = batch[node];
    const float4 v = *(const float4*)(h + (long long)node * HID + cg);
    float* pd = pool_s + b * HID + cg;
    atomicAdd(pd + 0, v.x * w);
    atomicAdd(pd + 1, v.y * w);
    atomicAdd(pd + 2, v.z * w);
    atomicAdd(pd + 3, v.w * w);
}

// ---------------------------------------------------------------------------
// MLP heads: phase_logits (64x3) then transition_prob (64x1) into d_out
// ---------------------------------------------------------------------------
__global__ void head_kernel(const float* __restrict__ pool_s, const float* __restrict__ bcnt,
                            const float* __restrict__ phW1, const float* __restrict__ phb1,
                            const float* __restrict__ phW2, const float* __restrict__ phb2,
                            const float* __restrict__ trW1, const float* __restrict__ trb1,
                            const float* __restrict__ trW2, const float* __restrict__ trb2,
                            float* __restrict__ out) {
    int b = threadIdx.x;
    if (b >= NB) return;
    float inv = 1.0f / fmaxf(bcnt[b], 1.0f);
    float p[HID];
#pragma unroll
    for (int c = 0; c < HID; ++c) p[c] = pool_s[b * HID + c] * inv;

    float h1[32];
    for (int j = 0; j < 32; ++j) {
        float a = phb1[j];
        for (int c = 0; c < HID; ++c) a += p[c] * phW1[c * 32 + j];
        h1[j] = fmaxf(a, 0.0f);
    }
    for (int j = 0; j < 3; ++j) {
        float a = phb2[j];
        for (int k = 0; k < 32; ++k) a += h1[k] * phW2[k * 3 + j];
        out[b * 3 + j] = a;
    }
    float h2[16];
    for (int j = 0; j < 16; ++j) {
        float a = trb1[j];
        for (int c = 0; c < HID; ++c) a += p[c] * trW1[c * 16 + j];
        h2[j] = fmaxf(a, 0.0f);
    }
    float a = trb2[0];
    for (int k = 0; k < 16; ++k) a += h2[k] * trW2[k];
    out[NB * 3 + b] = 1.0f / (1.0f + __expf(-a));
}

// ---------------------------------------------------------------------------
// Host launcher
// ---------------------------------------------------------------------------
extern "C" void kernel_launch(void* const* d_in, const int* in_sizes, int n_in,
                              void* d_out, int out_size, void* d_ws, size_t ws_size,
                              hipStream_t stream) {
    const float* x     = (const float*)d_in[0];
    const int*   ei    = (const int*)d_in[1];
    const int*   batch = (const int*)d_in[2];
    const float* W1l = (const float*)d_in[3];
    const float* b1  = (const float*)d_in[4];
    const float* W1r = (const float*)d_in[5];
    const float* W2l = (const float*)d_in[6];
    const float* b2  = (const float*)d_in[7];
    const float* W2r = (const float*)d_in[8];
    const float* W3l = (const float*)d_in[9];
    const float* b3  = (const float*)d_in[10];
    const float* W3r = (const float*)d_in[11];
    const float* g1  = (const float*)d_in[12];
    const float* be1 = (const float*)d_in[13];
    const float* g2  = (const float*)d_in[14];
    const float* be2 = (const float*)d_in[15];
    const float* g3  = (const float*)d_in[16];
    const float* be3 = (const float*)d_in[17];
    const float* phW1 = (const float*)d_in[18];
    const float* phb1 = (const float*)d_in[19];
    const float* phW2 = (const float*)d_in[20];
    const float* phb2 = (const float*)d_in[21];
    const float* trW1 = (const float*)d_in[22];
    const float* trb1 = (const float*)d_in[23];
    const float* trW2 = (const float*)d_in[24];
    const float* trb2 = (const float*)d_in[25];

    const int n = in_sizes[0] / 6;
    const int E = in_sizes[1] / 2;
    const int* src = ei;
    const int* dst = ei + E;

    // workspace carve-up (floats)
    float* ws = (float*)d_ws;
    size_t off = 0;
    float* deg    = ws + off; off += (size_t)n;
    float* agg    = ws + off; off += (size_t)n * HID;
    float* hA     = ws + off; off += (size_t)n * HID;
    float* hB     = ws + off; off += (size_t)n * HID;
    float* stats  = ws + off; off += 128;
    float* part   = ws + off; off += 512;
    float* md     = ws + off; off += 2;
    float* pool_s = ws + off; off += NB * HID;
    float* bcnt   = ws + off; off += NB;
    float* xpad   = ws + off; off += (size_t)n * 8;

    const int ntiles     = n / 16;
    const int gemmBlocks = (ntiles + 7) / 8;
    const long long e16  = (long long)E * 16;
    const unsigned egrid    = (unsigned)((E + 255) / 256);
    const unsigned egrid64  = (unsigned)((e16 + 255) / 256);
    const unsigned ngrid64  = (unsigned)(((long long)n * HID + 255) / 256);

    // degrees (shared by all 3 layers)
    hipMemsetAsync(deg, 0, (size_t)n * sizeof(float), stream);
    degree_kernel<<<egrid, 256, 0, stream>>>(dst, deg, E);

    // ---- layer 1 (K padded 6 -> 8, branch-free WMMA) ----
    pad6to8<<<(unsigned)(((long long)n * 8 + 255) / 256), 256, 0, stream>>>(x, xpad, n);
    hipMemsetAsync(agg, 0, (size_t)n * 8 * sizeof(float), stream);
    scatter_sum6p<<<egrid, 256, 0, stream>>>(x, src, dst, agg, E);
    hipMemsetAsync(stats, 0, 128 * sizeof(float), stream);
    sage_gemm_wmma<8, 6><<<gemmBlocks, 256, 0, stream>>>(agg, xpad, deg, W1l, W1r, b1, hA, stats, n);
    bn_relu_kernel<<<ngrid64, 256, 0, stream>>>(hA, stats, g1, be1, n);

    // ---- layer 2 (K = 64) ----
    hipMemsetAsync(agg, 0, (size_t)n * HID * sizeof(float), stream);
    scatter_sum64<<<egrid64, 256, 0, stream>>>(hA, src, dst, agg, e16);
    hipMemsetAsync(stats, 0, 128 * sizeof(float), stream);
    sage_gemm_wmma<64, 64><<<gemmBlocks, 256, 0, stream>>>(agg, hA, deg, W2l, W2r, b2, hB, stats, n);
    bn_relu_kernel<<<ngrid64, 256, 0, stream>>>(hB, stats, g2, be2, n);

    // ---- layer 3 (K = 64) ----
    hipMemsetAsync(agg, 0, (size_t)n * HID * sizeof(float), stream);
    scatter_sum64<<<egrid64, 256, 0, stream>>>(hB, src, dst, agg, e16);
    hipMemsetAsync(stats, 0, 128 * sizeof(float), stream);
    sage_gemm_wmma<64, 64><<<gemmBlocks, 256, 0, stream>>>(agg, hB, deg, W3l, W3r, b3, hA, stats, n);
    bn_relu_kernel<<<ngrid64, 256, 0, stream>>>(hA, stats, g3, be3, n);

    // ---- softmax over x[:,4] ----
    reduce_max_part<<<256, 256, 0, stream>>>(x, n, part);
    reduce_max_final<<<1, 256, 0, stream>>>(part, md);
    reduce_sumexp_part<<<256, 256, 0, stream>>>(x, md, n, part + 256);
    reduce_sumexp_final<<<1, 256, 0, stream>>>(part + 256, md);

    // ---- pooling ----
    hipMemsetAsync(pool_s, 0, NB * HID * sizeof(float), stream);
    hipMemsetAsync(bcnt, 0, NB * sizeof(float), stream);
    bcnt_kernel<<<(unsigned)((n + 255) / 256), 256, 0, stream>>>(batch, bcnt, n);
    pool_scatter<<<(unsigned)(((long long)n * 16 + 255) / 256), 256, 0, stream>>>(hA, x, batch, md, pool_s, n);

    // ---- heads ----
    head_kernel<<<1, 64, 0, stream>>>(pool_s, bcnt, phW1, phb1, phW2, phb2,
                                      trW1, trb1, trW2, trb2, (float*)d_out);
}